// CosineContrastiveLoss_19473381720389
// MI455X (gfx1250) — compile-verified
//
#include <hip/hip_runtime.h>
#include <hip/hip_bf16.h>
#include <math.h>

typedef __attribute__((ext_vector_type(2))) float v2f;
typedef __attribute__((ext_vector_type(8))) float v8f;

#define BN 4096
#define DD 256
#define TILES 256            // 4096 / 16 (M tiles)
#define GROUPS 64            // TILES / 4 (each wave does a 16x64 strip)
#define NPART (TILES * GROUPS)
#define NPAIRS 8386560.0f    // 4096*4095/2

// ---------------------------------------------------------------------------
// Kernel 1: per-row L2 norm + per-sample superclass id.
// One wave32 per row; lane sums 8 contiguous floats, then xor-shuffle reduce.
// ---------------------------------------------------------------------------
__global__ void __launch_bounds__(256)
prep_kernel(const float* __restrict__ P, const int* __restrict__ labels,
            const int* __restrict__ csid, float* __restrict__ norms,
            int* __restrict__ sup) {
  const int wave = blockIdx.x * (blockDim.x >> 5) + (threadIdx.x >> 5);
  const int lane = threadIdx.x & 31;
  if (wave >= BN) return;
  const float* row = P + (size_t)wave * DD + lane * 8;
  float s = 0.f;
#pragma unroll
  for (int i = 0; i < 8; ++i) { float v = row[i]; s += v * v; }
#pragma unroll
  for (int off = 16; off > 0; off >>= 1) s += __shfl_xor(s, off, 32);
  if (lane == 0) {
    norms[wave] = sqrtf(s);
    sup[wave]   = csid[labels[wave]];
  }
}

// ---------------------------------------------------------------------------
// Kernel 2: fused Gram tile (fp32 WMMA) + contrastive-loss epilogue.
// One wave computes a 16(M) x 64(N) strip: 4 accumulators share each A frag.
// VGPR layouts per CDNA5 ISA 7.12.2 (32-bit A 16x4, B 4x16, C/D 16x16):
//   A frag lane(l,h): v0 = A[l][k+2h], v1 = A[l][k+2h+1]   (float2 load)
//   B frag lane(l,h): v0 = B[2h][l],   v1 = B[2h+1][l]
//     for Gram, B[k][n] = P[n0+n][k]  ->  same float2 pattern, row n0+l.
//   C/D: component r of lane(l,h) = C[M = r + 8h][N = l].
// ---------------------------------------------------------------------------
__global__ void __launch_bounds__(32)
gram_loss_kernel(const float* __restrict__ P, const int* __restrict__ labels,
                 const float* __restrict__ norms, const int* __restrict__ sup,
                 float* __restrict__ partials) {
  const int g    = blockIdx.x;            // N group: bj tiles 4g .. 4g+3
  const int bi   = blockIdx.y;            // M tile
  const int lane = threadIdx.x & 31;
  const int pid  = bi * GROUPS + g;

  // Whole strip strictly below the diagonal: uniform (scalar) early-out.
  if (bi > 4 * g + 3) { if (lane == 0) partials[pid] = 0.f; return; }

  const int l  = lane & 15;
  const int h  = lane >> 4;
  const int m0 = bi * 16;
  const int n0 = g * 64;

  const float* pa  = P + (size_t)(m0 +      l) * DD + 2 * h;
  const float* pb0 = P + (size_t)(n0 +  0 + l) * DD + 2 * h;
  const float* pb1 = P + (size_t)(n0 + 16 + l) * DD + 2 * h;
  const float* pb2 = P + (size_t)(n0 + 32 + l) * DD + 2 * h;
  const float* pb3 = P + (size_t)(n0 + 48 + l) * DD + 2 * h;

  v8f c0 = {}, c1 = {}, c2 = {}, c3 = {};
#pragma unroll 4
  for (int k = 0; k < DD; k += 4) {
    v2f a  = *(const v2f*)(pa  + k);
    v2f b0 = *(const v2f*)(pb0 + k);
    v2f b1 = *(const v2f*)(pb1 + k);
    v2f b2 = *(const v2f*)(pb2 + k);
    v2f b3 = *(const v2f*)(pb3 + k);
    c0 = __builtin_amdgcn_wmma_f32_16x16x4_f32(false, a, false, b0, (short)0, c0, false, false);
    c1 = __builtin_amdgcn_wmma_f32_16x16x4_f32(false, a, false, b1, (short)0, c1, false, false);
    c2 = __builtin_amdgcn_wmma_f32_16x16x4_f32(false, a, false, b2, (short)0, c2, false, false);
    c3 = __builtin_amdgcn_wmma_f32_16x16x4_f32(false, a, false, b3, (short)0, c3, false, false);
  }

  // Row-side (M) metadata, shared across the four N sub-tiles.
  float nm[8]; int lm[8], sm[8];
#pragma unroll
  for (int r = 0; r < 8; ++r) {
    const int m = m0 + r + 8 * h;
    nm[r] = norms[m]; lm[r] = labels[m]; sm[r] = sup[m];
  }

  const v8f accs[4] = {c0, c1, c2, c3};
  float sum = 0.f;
#pragma unroll
  for (int t = 0; t < 4; ++t) {
    const int   n  = n0 + 16 * t + l;
    const float nn = norms[n];
    const int   ln = labels[n];
    const int   sn = sup[n];
#pragma unroll
    for (int r = 0; r < 8; ++r) {
      const int m = m0 + r + 8 * h;
      if (m < n) {                             // only unordered pairs i<j
        const float denom = fmaxf(nm[r] * nn, 1e-8f);
        const float dist  = 1.0f - accs[t][r] / denom;
        float v;
        if (lm[r] == ln) {
          const float x = fmaxf(dist - 0.05f, 0.0f); v = x * x;   // positive
        } else {
          const float mar = (sm[r] == sn) ? 0.3f : 0.5f;
          const float x = fmaxf(mar - dist, 0.0f); v = x * x;      // negative
        }
        sum += v;
      }
    }
  }
#pragma unroll
  for (int off = 16; off > 0; off >>= 1) sum += __shfl_xor(sum, off, 32);
  if (lane == 0) partials[pid] = sum;
}

// ---------------------------------------------------------------------------
// Kernel 3: deterministic fixed-order reduction of per-wave partials.
// ---------------------------------------------------------------------------
__global__ void __launch_bounds__(256)
reduce_kernel(const float* __restrict__ partials, float* __restrict__ out) {
  __shared__ float lds[256];
  const int t = threadIdx.x;
  float s = 0.f;
  for (int i = t; i < NPART; i += 256) s += partials[i];
  lds[t] = s;
  __syncthreads();
  for (int off = 128; off > 0; off >>= 1) {
    if (t < off) lds[t] += lds[t + off];
    __syncthreads();
  }
  if (t == 0) out[0] = lds[0] / NPAIRS;
}

// ---------------------------------------------------------------------------
extern "C" void kernel_launch(void* const* d_in, const int* in_sizes, int n_in,
                              void* d_out, int out_size, void* d_ws, size_t ws_size,
                              hipStream_t stream) {
  const float* P      = (const float*)d_in[0];
  const int*   labels = (const int*)d_in[1];
  const int*   csid   = (const int*)d_in[2];

  // Workspace layout: [norms: 4096 f32][sup: 4096 i32][partials: 16384 f32]
  float* norms    = (float*)d_ws;
  int*   sup      = (int*)((char*)d_ws + BN * sizeof(float));
  float* partials = (float*)((char*)d_ws + 2 * BN * sizeof(float));

  prep_kernel<<<BN / 8, 256, 0, stream>>>(P, labels, csid, norms, sup);

  dim3 grid(GROUPS, TILES);
  gram_loss_kernel<<<grid, 32, 0, stream>>>(P, labels, norms, sup, partials);

  reduce_kernel<<<1, 256, 0, stream>>>(partials, (float*)d_out);
}